// ShiftMap_74929999446167
// MI455X (gfx1250) — compile-verified
//
#include <hip/hip_runtime.h>
#include <hip/hip_bf16.h>

typedef float v2f __attribute__((ext_vector_type(2)));
typedef float v8f __attribute__((ext_vector_type(8)));

#define CUBIC_A (-0.75f)

__device__ __forceinline__ void cubic_weights(float t, float w[4]) {
    float t2 = t * t;
    float s  = 1.0f - t;
    w[1] = (CUBIC_A + 2.0f) * t * t2 - (CUBIC_A + 3.0f) * t2 + 1.0f;
    w[2] = (CUBIC_A + 2.0f) * s * s * s - (CUBIC_A + 3.0f) * s * s + 1.0f;
    float u = 1.0f + t;
    w[0] = CUBIC_A * (u * (u * (u - 5.0f) + 8.0f) - 4.0f);
    float v = 2.0f - t;
    w[3] = CUBIC_A * (v * (v * (v - 5.0f) + 8.0f) - 4.0f);
}

// ---------------------------------------------------------------------------
// Kernel 1: row-direction bicubic resize of the 16x16x2 shiftmap to 1024x16x2
// as a real fp32 matmul on the WMMA pipe:
//     tmpH(1024x16) = Wy(1024x16) @ SM(16x16)   per (batch, channel)
// One wave per 16-row tile; 4 chained V_WMMA_F32_16X16X4_F32 (K=16 total).
// tmpH layout: [b][y][ctrl_col][ch]  -> B*1024*16*2 floats (1 MB @ B=8).
// ---------------------------------------------------------------------------
__global__ void shiftmap_rows_wmma(const float* __restrict__ sm,
                                   float* __restrict__ tmpH)
{
    const int lane = threadIdx.x & 31;
    const int wave = threadIdx.x >> 5;
    const int job  = blockIdx.x * 8 + wave;   // B*2*64 jobs total, grid exact
    const int tile = job & 63;                // 64 tiles of 16 output rows
    const int ch   = (job >> 6) & 1;
    const int b    = job >> 7;

    const int m   = lane & 15;                // M index (both lane halves)
    const int row = tile * 16 + m;            // output row 0..1023
    const int kh  = (lane >= 16) ? 2 : 0;     // K split: upper lanes hold K+2

    // cubic row weights: src = row * (cin-1)/(pout-1), taps at i0-1..i0+2 clamped
    float s = (float)row * (15.0f / 1023.0f);
    float f = floorf(s);
    float t = s - f;
    int   i0 = (int)f;
    float w[4];
    cubic_weights(t, w);
    const int c0 = min(max(i0 - 1, 0), 15);
    const int c1 = min(max(i0 + 0, 0), 15);
    const int c2 = min(max(i0 + 1, 0), 15);
    const int c3 = min(max(i0 + 2, 0), 15);

    const float* smb = sm + (size_t)b * 512 + ch;   // (16,16,2) for this batch

    v8f acc = {};
    #pragma unroll
    for (int kc = 0; kc < 4; ++kc) {
        const int ka  = 4 * kc + kh;   // K index for VGPR0 of A / row of B
        const int ka1 = ka + 1;        // K index for VGPR1
        v2f a, bm;
        // dense Wy[row][ka], Wy[row][ka1]: sum of the (possibly clamped) taps
        a.x = (ka  == c0 ? w[0] : 0.f) + (ka  == c1 ? w[1] : 0.f) +
              (ka  == c2 ? w[2] : 0.f) + (ka  == c3 ? w[3] : 0.f);
        a.y = (ka1 == c0 ? w[0] : 0.f) + (ka1 == c1 ? w[1] : 0.f) +
              (ka1 == c2 ? w[2] : 0.f) + (ka1 == c3 ? w[3] : 0.f);
        // B[k][n] = SM[ctrl_row=k][ctrl_col=n][ch]
        const int n = lane & 15;
        bm.x = smb[(ka  * 16 + n) * 2];
        bm.y = smb[(ka1 * 16 + n) * 2];
        acc = __builtin_amdgcn_wmma_f32_16x16x4_f32(
                  false, a, false, bm, (short)0, acc, false, false);
    }

    // D layout: VGPR r, lanes 0-15 -> M=r, lanes 16-31 -> M=r+8; N = lane&15
    const int n    = lane & 15;
    const int mofs = (lane >= 16) ? 8 : 0;
    #pragma unroll
    for (int r = 0; r < 8; ++r) {
        const int mm = tile * 16 + r + mofs;
        tmpH[(((size_t)b * 1024) + mm) * 32 + n * 2 + ch] = acc[r];
    }
}

// ---------------------------------------------------------------------------
// Kernel 2: fused column-direction shiftmap resize + mesh add + bicubic
// grid_sample (zeros padding, align_corners=True). One thread per output
// pixel; 32x8 tiles so the 16 image taps of a block share cache lines.
// ---------------------------------------------------------------------------
__global__ void warp_bicubic(const float* __restrict__ img,
                             const float2* __restrict__ mesh,
                             const float* __restrict__ tmpH,
                             float* __restrict__ out)
{
    const int H = 1024, W = 1024;
    const int x = blockIdx.x * 32 + (threadIdx.x & 31);
    const int y = blockIdx.y * 8  + (threadIdx.x >> 5);
    const int b = blockIdx.z;

    const float2 mv = mesh[(size_t)y * W + x];   // (.x = mesh_y, .y = mesh_x)

    // column-direction bicubic of the row-resized control data
    float sx = (float)x * (15.0f / 1023.0f);
    float fx = floorf(sx);
    int  jx0 = (int)fx;
    float wc[4];
    cubic_weights(sx - fx, wc);
    const float2* th = (const float2*)(tmpH + (((size_t)b * 1024) + y) * 32);
    float smy = 0.f, smx = 0.f;
    #pragma unroll
    for (int k = 0; k < 4; ++k) {
        const int c = min(max(jx0 - 1 + k, 0), 15);
        const float2 v = th[c];
        smy = fmaf(wc[k], v.x, smy);
        smx = fmaf(wc[k], v.y, smx);
    }

    const float gy = mv.x + smy;
    const float gx = mv.y + smx;
    const float ixf = (gx + 1.0f) * 0.5f * (float)(W - 1);
    const float iyf = (gy + 1.0f) * 0.5f * (float)(H - 1);
    const float fix = floorf(ixf);
    const float fiy = floorf(iyf);
    const int ix0 = (int)fix;
    const int iy0 = (int)fiy;
    float wx[4], wy[4];
    cubic_weights(ixf - fix, wx);
    cubic_weights(iyf - fiy, wy);

    const float* imgb = img + (size_t)b * H * W;
    float acc = 0.f;
    #pragma unroll
    for (int i = 0; i < 4; ++i) {
        const int yy  = iy0 - 1 + i;
        const int yc  = min(max(yy, 0), H - 1);
        const float wyi = (yy >= 0 && yy < H) ? wy[i] : 0.f;   // zeros padding
        const float* rowp = imgb + (size_t)yc * W;
        #pragma unroll
        for (int j = 0; j < 4; ++j) {
            const int xx = ix0 - 1 + j;
            const int xc = min(max(xx, 0), W - 1);
            const float wgt = (xx >= 0 && xx < W) ? wyi * wx[j] : 0.f;
            acc = fmaf(wgt, rowp[xc], acc);
        }
    }
    out[(((size_t)b * H) + y) * W + x] = acc;
}

extern "C" void kernel_launch(void* const* d_in, const int* in_sizes, int n_in,
                              void* d_out, int out_size, void* d_ws, size_t ws_size,
                              hipStream_t stream)
{
    const int H = 1024, W = 1024;
    const int B = in_sizes[0] / (H * W);          // 8
    const float*  img  = (const float*)d_in[0];   // (B,1,H,W)
    const float*  smap = (const float*)d_in[1];   // (B,16,16,2)
    const float2* mesh = (const float2*)d_in[2];  // (1,H,W,2)
    float* out  = (float*)d_out;                  // (B,1,H,W)
    float* tmpH = (float*)d_ws;                   // B*1024*16*2 f32 = 1 MB

    // one wave per 16-row tile: B*2*64 jobs, 8 waves/block -> EXEC all-ones
    shiftmap_rows_wmma<<<dim3(B * 16), 256, 0, stream>>>(smap, tmpH);
    warp_bicubic<<<dim3(W / 32, H / 8, B), 256, 0, stream>>>(img, mesh, tmpH, out);
}